// CARAFEUpsample_19559281066037
// MI455X (gfx1250) — compile-verified
//
#include <hip/hip_runtime.h>
#include <hip/hip_bf16.h>

typedef __attribute__((ext_vector_type(2))) float v2f;
typedef __attribute__((ext_vector_type(8))) float v8f;

#define EPS 1e-5f

// ---------------------------------------------------------------------------
// Kernel 0: weight repack (runs once per launch, trivially cheap).
// Interleaved pair layout so each lane's (k, k+1) A-operands are one b64 load:
//   wCw[(k>>1)*128 + o*2 + (k&1)]   = comp_w[o*128 + k]          (128k x 64o)
//   wAe[(t>>1)*224 + o*2 + (t&1)]   = enc_w[o*576 + c*9 + off]   (576k x 112o,
//                                      t = off*64 + c, o padded 100->112)
// ---------------------------------------------------------------------------
__global__ void k_repack(const float* __restrict__ comp_w,
                         const float* __restrict__ enc_w,
                         float* __restrict__ wCw,
                         float* __restrict__ wAe) {
    const int i = blockIdx.x * blockDim.x + threadIdx.x;
    if (i < 8192) {                        // pair*128 + o*2 + parity
        const int o = (i >> 1) & 63;
        const int k = ((i >> 7) << 1) | (i & 1);
        wCw[i] = comp_w[o * 128 + k];
    }
    if (i < 64512) {                       // pair*224 + o*2 + parity
        const int o = (i >> 1) % 112;
        const int t = ((i / 224) << 1) | (i & 1);
        const int c = t & 63, off = t >> 6;
        wAe[i] = (o < 100) ? enc_w[(size_t)o * 576 + c * 9 + off] : 0.f;
    }
}

// ---------------------------------------------------------------------------
// Kernel 1: channel compressor. y = SiLU(BN(x @ comp_w))
// Block = one 16-px spatial tile; 4 waves = the 4 o-tiles sharing the
// B tile (128ch x 16px, 8KB) staged once in LDS. A loads are b64.
// ---------------------------------------------------------------------------
__global__ void k_compress(const float* __restrict__ x,
                           const float* __restrict__ wCw,
                           const float* __restrict__ gamma,
                           const float* __restrict__ beta,
                           const float* __restrict__ mean,
                           const float* __restrict__ var,
                           float* __restrict__ y) {
    __shared__ float ls[128 * 16];
    const int tid    = threadIdx.x;
    const int sTile  = blockIdx.x;
    const int b      = sTile >> 8;
    const int hwBase = (sTile & 255) << 4;

    const float* xb = x + (size_t)b * 128 * 4096;
#pragma unroll
    for (int i = tid; i < 2048; i += 128)          // coalesced stage
        ls[i] = xb[(size_t)(i >> 4) * 4096 + hwBase + (i & 15)];
    __syncthreads();

    const int lane = tid & 31, l = lane & 15, half = lane >> 4;
    const int oBase = (tid >> 5) << 4;             // wave id == o tile
    const int aCol = (oBase + l) * 2;

    v8f acc = {};
#pragma unroll 4
    for (int kb = 0; kb < 128; kb += 4) {
        const int k0 = kb + half * 2;
        const v2f a = *(const v2f*)(wCw + ((kb >> 1) + half) * 128 + aCol);
        v2f bm;
        bm.x = ls[k0 * 16 + l];
        bm.y = ls[(k0 + 1) * 16 + l];
        acc = __builtin_amdgcn_wmma_f32_16x16x4_f32(
            false, a, false, bm, (short)0, acc, false, false);
    }

#pragma unroll
    for (int j = 0; j < 8; ++j) {
        const int o = oBase + j + half * 8;        // C/D row layout
        const float iv = gamma[o] * rsqrtf(var[o] + EPS);
        const float bb = beta[o] - mean[o] * iv;
        float v = acc[j] * iv + bb;
        v = v * __frcp_rn(1.f + __expf(-v));       // SiLU
        y[((size_t)b * 64 + o) * 4096 + hwBase + l] = v;
    }
}

// ---------------------------------------------------------------------------
// Kernel 2: content encoder, 3x3 SAME conv 64->100(+bias), K = 9*64 = 576.
// Block = one spatial tile, 7 waves = 7 o-tiles. The 3x18x64ch input halo
// patch (13.8KB) is staged in LDS with OOB zeros -> branch-free WMMA loop.
// A loads are b64 from the interleaved repack.
// ---------------------------------------------------------------------------
__global__ void k_encconv(const float* __restrict__ y,
                          const float* __restrict__ wAe,
                          const float* __restrict__ enc_b,
                          float* __restrict__ logits) {
    __shared__ float ls[64 * 54];                  // [c][r(3)][col(18)]
    const int tid    = threadIdx.x;                // 0..223 (7 waves)
    const int sTile  = blockIdx.x;
    const int b      = sTile >> 8;
    const int hwBase = (sTile & 255) << 4;
    const int h      = hwBase >> 6;
    const int w0     = hwBase & 63;

    const float* yb = y + (size_t)b * 64 * 4096;
    for (int i = tid; i < 3456; i += 224) {
        const int c = i / 54, rem = i % 54;
        const int r = rem / 18, col = rem % 18;
        const int hh = h + r - 1, ww = w0 + col - 1;
        const bool ok = (hh >= 0) & (hh < 64) & (ww >= 0) & (ww < 64);
        ls[i] = ok ? yb[(size_t)c * 4096 + hh * 64 + ww] : 0.f;
    }
    __syncthreads();

    const int lane = tid & 31, l = lane & 15, half = lane >> 4;
    const int oBase = (tid >> 5) << 4;             // wave id == o tile (0..6)
    const int aCol = (oBase + l) * 2;

    v8f acc = {};
    for (int off = 0; off < 9; ++off) {
        const int ki = off / 3, kj = off % 3;
        const int bcol = ki * 18 + kj + l;         // LDS col for this lane
#pragma unroll 4
        for (int cb = 0; cb < 64; cb += 4) {
            const int c0 = cb + half * 2;
            const int pair = off * 32 + (cb >> 1) + half;
            const v2f a = *(const v2f*)(wAe + pair * 224 + aCol);
            v2f bm;
            bm.x = ls[c0 * 54 + bcol];
            bm.y = ls[(c0 + 1) * 54 + bcol];
            acc = __builtin_amdgcn_wmma_f32_16x16x4_f32(
                false, a, false, bm, (short)0, acc, false, false);
        }
    }

#pragma unroll
    for (int j = 0; j < 8; ++j) {
        const int oo = oBase + j + half * 8;
        if (oo < 100)
            logits[((size_t)b * 100 + oo) * 4096 + hwBase + l] = acc[j] + enc_b[oo];
    }
}

// ---------------------------------------------------------------------------
// Kernel 3: pixel-shuffle + softmax(25) + fused CARAFE reassembly.
// The 25 taps on the upsampled grid fold onto a 3x3 source-grid patch:
//   src row of tap ki = (Y-2)>>1 + ((ki + (Y&1)) >> 1)  (same for cols),
// and padded taps land exactly on OOB src cells. Output rows Y=2t and
// Y=2t+1 need the SAME 3 source rows (hbase = t-1), so each 256-thread
// block computes a Y-pair from one 101KB LDS stage (OOB pre-zeroed):
// 9 immediate-offset ds_loads per channel, stage math hoisted per-thread.
// ---------------------------------------------------------------------------
__global__ void k_reassemble(const float* __restrict__ x,
                             const float* __restrict__ logits,
                             float* __restrict__ out) {
    extern __shared__ float ls[];                  // [c][r(3)][col(66)]
    const int tid = threadIdx.x;                   // 0..255
    const int X   = tid & 127;
    const int s1  = tid >> 7;                      // which Y of the pair
    const int b   = blockIdx.x >> 6;
    const int Yp  = blockIdx.x & 63;               // Y pair index (= h = Y>>1)
    const int Y   = Yp * 2 + s1;
    const int s2  = X & 1;
    const int hbase = Yp - 1;                      // floor((Y-2)/2), both rows

    // ---- stage x patch: rows hbase..hbase+2, cols -1..64 (zeros if OOB)
    const int r   = (tid < 198) ? tid / 66 : 0;    // per-thread constants
    const int col = (tid < 198) ? tid % 66 : 0;
    const int hh  = hbase + r, ww = col - 1;
    const bool ok = (tid < 198) & (hh >= 0) & (hh < 64) & (ww >= 0) & (ww < 64);
    const float* xb = x + (size_t)b * 128 * 4096;
    const int xoff  = hh * 64 + ww;
    if (tid < 198) {
#pragma unroll 4
        for (int c = 0; c < 128; ++c)
            ls[c * 198 + tid] = ok ? xb[(size_t)c * 4096 + xoff] : 0.f;
    }
    __syncthreads();

    // ---- softmax over the 25 mask logits of this shuffled sub-position
    const int w = X >> 1;
    const float* lg = logits + (size_t)b * 100 * 4096 + (size_t)Yp * 64 + w;
    float p[25];
    float mx = -1e30f;
#pragma unroll
    for (int q = 0; q < 25; ++q) {
        p[q] = lg[(size_t)(q * 4 + s1 * 2 + s2) * 4096];
        mx = fmaxf(mx, p[q]);
    }
    float sum = 0.f;
#pragma unroll
    for (int q = 0; q < 25; ++q) { p[q] = __expf(p[q] - mx); sum += p[q]; }
    const float rs = __frcp_rn(sum);

    // ---- fold 25 weights into 3x3 source-grid weights (OOB cells hold 0 data)
    float wc[9] = {0.f, 0.f, 0.f, 0.f, 0.f, 0.f, 0.f, 0.f, 0.f};
#pragma unroll
    for (int ki = 0; ki < 5; ++ki) {
        const int di = (ki + s1) >> 1;
#pragma unroll
        for (int kj = 0; kj < 5; ++kj) {
            const int dj = (kj + s2) >> 1;
            wc[di * 3 + dj] += p[ki * 5 + kj] * rs;
        }
    }

    // ---- 9-tap reassembly per channel from LDS
    const int base = ((X - 2) >> 1) + 1;           // LDS col of dj==0 (>= 0)
    float* ob = out + (size_t)b * 128 * 16384 + (size_t)Y * 128 + X;
#pragma unroll 2
    for (int c = 0; c < 128; ++c) {
        const float* lc = ls + c * 198 + base;
        float s = 0.f;
#pragma unroll
        for (int t = 0; t < 9; ++t)
            s += wc[t] * lc[(t / 3) * 66 + (t % 3)];
        ob[(size_t)c * 16384] = s;
    }
}

// ---------------------------------------------------------------------------
extern "C" void kernel_launch(void* const* d_in, const int* in_sizes, int n_in,
                              void* d_out, int out_size, void* d_ws, size_t ws_size,
                              hipStream_t stream) {
    const float* x      = (const float*)d_in[0];
    const float* comp_w = (const float*)d_in[1];
    const float* gamma  = (const float*)d_in[2];
    const float* beta   = (const float*)d_in[3];
    const float* mean   = (const float*)d_in[4];
    const float* var    = (const float*)d_in[5];
    const float* enc_w  = (const float*)d_in[6];
    const float* enc_b  = (const float*)d_in[7];
    float* out = (float*)d_out;

    float* y      = (float*)d_ws;                    // 1,048,576 f = 4   MB
    float* logits = y + (size_t)1048576;             // 1,638,400 f = 6.5 MB
    float* wCw    = logits + (size_t)1638400;        //     8,192 f (8B aligned)
    float* wAe    = wCw + (size_t)8192;              //    64,512 f (8B aligned)

    k_repack<<<252, 256, 0, stream>>>(comp_w, enc_w, wCw, wAe);
    k_compress<<<1024, 128, 0, stream>>>(x, wCw, gamma, beta, mean, var, y);
    k_encconv<<<1024, 224, 0, stream>>>(y, wAe, enc_b, logits);
    // 256 blocks = 4 b * 64 Y-pairs, 256 threads = 2 Y rows * 128 X
    k_reassemble<<<256, 256, 198 * 128 * sizeof(float), stream>>>(x, logits, out);
}